// Ranker_77446850282051
// MI455X (gfx1250) — compile-verified
//
#include <hip/hip_runtime.h>
#include <hip/hip_bf16.h>

#define B_ROWS 256
#define V_DIM  128000
#define L_DIM  200

typedef __attribute__((ext_vector_type(2))) float v2f;
typedef __attribute__((ext_vector_type(8))) float v8f;

// ---------------------------------------------------------------------------
// Kernel 1: one workgroup per row. Streams the 128000-score row with b128
// loads (memory-bound; whole 131MB problem fits in the 192MB L2), counts
// scores > predict, and corrects for masked (seqs) positions using a 16KB
// LDS bitset purely for duplicate detection (atomicOr old-value test).
// ---------------------------------------------------------------------------
__global__ __launch_bounds__(256) void row_rank_kernel(
    const float* __restrict__ scores,
    const int*   __restrict__ labels,
    const int*   __restrict__ seqs,
    int* __restrict__ rank_out,
    int* __restrict__ valid_out)
{
    __shared__ unsigned int bits[V_DIM / 32];   // 4000 dwords = 16000 B
    __shared__ int s_cnt;
    __shared__ int s_corr;
    __shared__ int s_dist;

    const int b   = blockIdx.x;
    const int tid = threadIdx.x;

    // zero the dedupe bitset + counters
    for (int i = tid; i < V_DIM / 32; i += 256) bits[i] = 0u;
    if (tid == 0) { s_cnt = 0; s_corr = 0; s_dist = 0; }
    __syncthreads();

    const int   label   = labels[b];                       // uniform -> SMEM
    const float predict = scores[(size_t)b * V_DIM + label];

    // Phase 1: distinct masked positions + correction count (threads 0..199)
    int corr = 0, dist = 0;
    if (tid < L_DIM) {
        const int v = seqs[b * L_DIM + tid];
        const unsigned int m   = 1u << (v & 31);
        const unsigned int old = atomicOr(&bits[v >> 5], m);
        if (!(old & m)) {                                   // first setter owns v
            dist = 1;
            if (scores[(size_t)b * V_DIM + v] > predict) corr = 1;
        }
    }

    // Phase 2: raw strict-greater count over the full row (b128 coalesced)
    const float4* __restrict__ row = (const float4*)(scores + (size_t)b * V_DIM);
    int cnt = 0;
#pragma unroll 5
    for (int i = 0; i < (V_DIM / 4) / 256; ++i) {           // 125 iterations
        const float4 f = row[i * 256 + tid];
        cnt += (f.x > predict) + (f.y > predict) + (f.z > predict) + (f.w > predict);
    }

    atomicAdd(&s_cnt, cnt);                                 // integer -> deterministic
    if (tid < L_DIM) {
        if (corr) atomicAdd(&s_corr, 1);
        if (dist) atomicAdd(&s_dist, 1);
    }
    __syncthreads();

    if (tid == 0) {
        rank_out[b]  = s_cnt - s_corr;    // rank over masked row
        valid_out[b] = V_DIM - s_dist;    // #unmasked entries
    }
}

// ---------------------------------------------------------------------------
// Kernel 2: per-row 13-metric vectors -> LDS table[256][16], then wave 0
// reduces all 256 rows with 64 chained V_WMMA_F32_16X16X4_F32 ops
// (A = ones 16x4, B = 4 rows x 16 metric columns per step). Full-f32,
// fixed-order, deterministic; D replicates each column sum across rows.
// ---------------------------------------------------------------------------
__global__ __launch_bounds__(256) void finalize_kernel(
    const int* __restrict__ rank_arr,
    const int* __restrict__ valid_arr,
    float* __restrict__ out)
{
    __shared__ float table[B_ROWS][16];

    const int t = threadIdx.x;
    const float rank   = (float)rank_arr[t];
    const float valid  = (float)valid_arr[t];
    const float invlog = 1.0f / log2f(rank + 2.0f);

    const float ks[5] = {1.0f, 5.0f, 10.0f, 20.0f, 50.0f};
    float m[16];
#pragma unroll
    for (int i = 0; i < 5; ++i) {
        const float ind = (rank < ks[i]) ? 1.0f : 0.0f;
        m[2 * i]     = ind * invlog;   // NDCG@k contribution
        m[2 * i + 1] = ind;            // Recall@k contribution
    }
    m[10] = 1.0f / (rank + 1.0f);      // MRR
    m[11] = 1.0f - rank / valid;       // AUC-like
    m[12] = 0.0f;                      // literal 0 output
    m[13] = 0.0f; m[14] = 0.0f; m[15] = 0.0f;  // pad columns

#pragma unroll
    for (int c = 0; c < 16; ++c) table[t][c] = m[c];
    __syncthreads();

    if (t < 32) {                      // whole wave 0: EXEC all ones for WMMA
        const int lane = t;
        const int col  = lane & 15;    // metric column
        const int kh   = lane >> 4;    // K half select (lanes 16-31 -> K=2,3)

        v8f acc = {};
        v2f a;  a.x = 1.0f; a.y = 1.0f;   // all-ones A (16x4)

        for (int i = 0; i < 64; ++i) {    // 64 * 4 rows = 256 rows
            v2f bv;
            bv.x = table[4 * i + 2 * kh + 0][col];
            bv.y = table[4 * i + 2 * kh + 1][col];
            acc = __builtin_amdgcn_wmma_f32_16x16x4_f32(
                /*neg_a=*/false, a, /*neg_b=*/false, bv,
                /*c_mod=*/(short)0, acc, /*reuse_a=*/false, /*reuse_b=*/false);
        }
        // every D row holds the column sum; lane<13 reads its column's sum
        if (lane < 13) out[lane] = acc.s0 * (1.0f / (float)B_ROWS);
    }
}

// ---------------------------------------------------------------------------
extern "C" void kernel_launch(void* const* d_in, const int* in_sizes, int n_in,
                              void* d_out, int out_size, void* d_ws, size_t ws_size,
                              hipStream_t stream)
{
    const float* scores = (const float*)d_in[0];   // [256, 128000] f32
    const int*   labels = (const int*)  d_in[1];   // [256]
    const int*   seqs   = (const int*)  d_in[2];   // [256, 200]
    float*       out    = (float*)d_out;           // 13 f32

    int* rank_arr  = (int*)d_ws;
    int* valid_arr = rank_arr + B_ROWS;

    row_rank_kernel<<<B_ROWS, 256, 0, stream>>>(scores, labels, seqs,
                                                rank_arr, valid_arr);
    finalize_kernel<<<1, 256, 0, stream>>>(rank_arr, valid_arr, out);
}